// BikeFlowGNN_44873818308834
// MI455X (gfx1250) — compile-verified
//
#include <hip/hip_runtime.h>
#include <hip/hip_bf16.h>

#define NN 50000
#define EE 1600000
#define PP 100000
#define HH 128
#define FIN 32
#define LL 3
#define EPSV 1e-5f

typedef __attribute__((ext_vector_type(2))) float v2f;
typedef __attribute__((ext_vector_type(8))) float v8f;
typedef __attribute__((ext_vector_type(4))) int v4i;

// ---------- async global->LDS staging (gfx1250 ASYNCcnt path), with fallback ----------
#if defined(__has_builtin)
#if __has_builtin(__builtin_amdgcn_global_load_async_to_lds_b128) && \
    __has_builtin(__builtin_amdgcn_s_wait_asynccnt)
#define ASYNC_LDS 1
#endif
#endif
#ifndef ASYNC_LDS
#define ASYNC_LDS 0
#endif

// builtin signature (from clang diagnostic): param0 = int4 in AS1 (global src),
// param1 = int4 in AS3 (LDS dst), then imm offset, imm cpol
typedef __attribute__((address_space(1))) v4i* gv4i_p;
typedef __attribute__((address_space(3))) v4i* lv4i_p;

__device__ __forceinline__ void stage_f4(const float* __restrict__ g, float* l) {
#if ASYNC_LDS
    __builtin_amdgcn_global_load_async_to_lds_b128(
        (gv4i_p)(uintptr_t)g, (lv4i_p)(unsigned)(uintptr_t)l, 0, 0);
#else
    *(float4*)l = *(const float4*)g;
#endif
}
__device__ __forceinline__ void stage_join() {
#if ASYNC_LDS
    __builtin_amdgcn_s_wait_asynccnt(0);
#endif
}

// ---------------- degree / normalization ----------------
__global__ void deg_kernel(const int* __restrict__ dst, float* __restrict__ deg) {
    int e = blockIdx.x * blockDim.x + threadIdx.x;
    if (e < EE) atomicAdd(&deg[dst[e]], 1.0f);
}

__global__ void dinv_kernel(const float* __restrict__ deg, float* __restrict__ dinv,
                            float* __restrict__ nself) {
    int n = blockIdx.x * blockDim.x + threadIdx.x;
    if (n < NN) {
        float d = rsqrtf(deg[n] + 1.0f);
        dinv[n] = d;
        nself[n] = d * d;
    }
}

__global__ void norme_kernel(const int* __restrict__ src, const int* __restrict__ dst,
                             const float* __restrict__ dinv, float* __restrict__ ne) {
    int e = blockIdx.x * blockDim.x + threadIdx.x;
    if (e < EE) ne[e] = dinv[src[e]] * dinv[dst[e]];
}

// ---------------- generic fp32 WMMA GEMM (compile-time K, NCOLS) ----------------
// C[M x NCOLS] = act(A[M x K] * B[K x NCOLS] + bias)
// block = NCOLS*2 threads (NCOLS/16 waves); grid.x = M/16 (M multiple of 16)
// dynamic LDS = (16*(K+4) + K*(NCOLS+8)) floats
// LDS padding: A row stride K+4 (== 4*odd mod 64 banks -> 16 lanes hit 16 banks),
//              B row stride NCOLS+8 (half-wave row distance 2*stride == 16 mod 64).
template <int K, int NCOLS>
__global__ void gemm_wmma(const float* __restrict__ A, const float* __restrict__ B,
                          const float* __restrict__ bias, float* __restrict__ C,
                          int lda, int ldc, int act) {
    constexpr int SA = K + 4;
    constexpr int SB = NCOLS + 8;
    constexpr int NT = NCOLS * 2;  // threads per block
    extern __shared__ float lds[];
    float* ldsA = lds;             // 16 x SA
    float* ldsB = lds + 16 * SA;   // K x SB

    const int tid  = threadIdx.x;
    const int row0 = blockIdx.x * 16;

    // stage A tile: 16 rows x K floats, vec4, coalesced
    for (int i = tid; i < 16 * (K / 4); i += NT) {
        int r = i / (K / 4), c4 = i % (K / 4);
        stage_f4(&A[(size_t)(row0 + r) * lda + c4 * 4], &ldsA[r * SA + c4 * 4]);
    }
    // stage B panel: K rows x NCOLS floats
    for (int i = tid; i < K * (NCOLS / 4); i += NT) {
        int r = i / (NCOLS / 4), c4 = i % (NCOLS / 4);
        stage_f4(&B[(size_t)r * NCOLS + c4 * 4], &ldsB[r * SB + c4 * 4]);
    }
    stage_join();
    __syncthreads();

    const int lane = tid & 31;
    const int wv   = tid >> 5;
    const int n0   = wv << 4;
    const int l15  = lane & 15;
    const int kh   = (lane >> 4) << 1;  // 0 / 2 per half-wave (A: K-split, B: row-split)

    v8f acc = {};
#pragma unroll
    for (int kk = 0; kk < K; kk += 4) {
        v2f a;
        a.x = ldsA[l15 * SA + kk + kh];
        a.y = ldsA[l15 * SA + kk + kh + 1];
        v2f b;
        b.x = ldsB[(kk + kh) * SB + n0 + l15];
        b.y = ldsB[(kk + kh + 1) * SB + n0 + l15];
        acc = __builtin_amdgcn_wmma_f32_16x16x4_f32(false, a, false, b,
                                                    (short)0, acc, false, false);
    }

    const int n  = n0 + l15;
    const int mb = row0 + ((lane >> 4) << 3);
    const float bv = bias ? bias[n] : 0.0f;
#pragma unroll
    for (int r = 0; r < 8; ++r) {
        float v = acc[r] + bv;
        if (act) v = fmaxf(v, 0.0f);
        C[(size_t)(mb + r) * ldc + n] = v;
    }
}

// ---------------- hbuf init: hb = z * norm_self + bs ----------------
__global__ void init_h(const float* __restrict__ z, const float* __restrict__ nself,
                       const float* __restrict__ bsi, float* __restrict__ hb) {
    int idx = blockIdx.x * blockDim.x + threadIdx.x;  // N*H threads
    int n = idx >> 7, j = idx & 127;
    hb[idx] = z[idx] * nself[n] + bsi[j];
}

// ---------------- edge scatter: one wave per edge, float4 per lane ----------------
__global__ void edge_agg(const int* __restrict__ src, const int* __restrict__ dst,
                         const float* __restrict__ ne, const float* __restrict__ z,
                         float* __restrict__ hb) {
    int gid  = blockIdx.x * blockDim.x + threadIdx.x;
    int e    = gid >> 5;
    int lane = gid & 31;
    if (e < EE) {
        int s = src[e], d = dst[e];
        float w = ne[e];
        const float4 zv = *(const float4*)(z + (size_t)s * HH + lane * 4);
        float* p = hb + (size_t)d * HH + lane * 4;
        atomicAdd(p + 0, zv.x * w);
        atomicAdd(p + 1, zv.y * w);
        atomicAdd(p + 2, zv.z * w);
        atomicAdd(p + 3, zv.w * w);
    }
}

// ---------------- batch-norm ----------------
__global__ void bn_stats(const float* __restrict__ hb, float* __restrict__ sums,
                         float* __restrict__ sqs) {
    int j = threadIdx.x & 127;
    int rstart = blockIdx.x * (blockDim.x >> 7) + (threadIdx.x >> 7);
    int stride = gridDim.x * (blockDim.x >> 7);
    float s = 0.0f, q = 0.0f;
    for (int r = rstart; r < NN; r += stride) {
        float v = hb[(size_t)r * HH + j];
        s += v;
        q += v * v;
    }
    atomicAdd(&sums[j], s);
    atomicAdd(&sqs[j], q);
}

__global__ void bn_final(const float* __restrict__ sums, const float* __restrict__ sqs,
                         float* __restrict__ mu, float* __restrict__ rstd) {
    int j = threadIdx.x;
    float m = sums[j] * (1.0f / (float)NN);
    float v = sqs[j] * (1.0f / (float)NN) - m * m;
    mu[j] = m;
    rstd[j] = rsqrtf(v + EPSV);
}

__global__ void bn_apply(const float* __restrict__ hb, const float* __restrict__ mu,
                         const float* __restrict__ rstd, const float* __restrict__ gam,
                         const float* __restrict__ bet, int relu, float* __restrict__ h) {
    int idx = blockIdx.x * blockDim.x + threadIdx.x;  // N*H threads
    int j = idx & 127;
    float v = (hb[idx] - mu[j]) * rstd[j] * gam[j] + bet[j];
    if (relu) v = fmaxf(v, 0.0f);
    h[idx] = v;
}

// ---------------- pair MLP layer 1 ----------------
// gathers h[src]||h[dst] (K=256) into padded LDS, stages W1[0:256,:] into padded LDS,
// WMMA over K=256, epilogue folds time-feature rows (256,257) + bias + relu.
// block = 256 threads; grid = P/16; dynamic LDS = (16*260 + 256*136) floats
__global__ void pair_mlp1_wmma(const float* __restrict__ h, const int* __restrict__ sn,
                               const int* __restrict__ tn, const float* __restrict__ tf,
                               const float* __restrict__ W1, const float* __restrict__ b1,
                               float* __restrict__ y1) {
    constexpr int SA = 260;  // 256+4
    constexpr int SB = 136;  // 128+8
    extern __shared__ float lds[];
    float* ldsA = lds;              // 16 x SA
    float* ldsB = lds + 16 * SA;    // 256 x SB

    const int tid  = threadIdx.x;
    const int row0 = blockIdx.x * 16;

    // A tile: 16 rows x 64 vec4 (32 from h[sn], 32 from h[tn]); each wave of 32
    // consecutive i takes one uniform branch (q entirely <32 or >=32) -> no divergence
    for (int i = tid; i < 16 * 64; i += 256) {
        int r = i >> 6, q = i & 63;
        if (q < 32)
            stage_f4(&h[(size_t)sn[row0 + r] * HH + q * 4], &ldsA[r * SA + q * 4]);
        else
            stage_f4(&h[(size_t)tn[row0 + r] * HH + (q - 32) * 4],
                     &ldsA[r * SA + 128 + (q - 32) * 4]);
    }
    // W1 panel: 256 rows x 32 vec4
    for (int i = tid; i < 256 * 32; i += 256) {
        int r = i >> 5, c4 = i & 31;
        stage_f4(&W1[(size_t)r * HH + c4 * 4], &ldsB[r * SB + c4 * 4]);
    }
    stage_join();
    __syncthreads();

    const int lane = tid & 31;
    const int wv   = tid >> 5;
    const int n0   = wv << 4;
    const int l15  = lane & 15;
    const int kh   = (lane >> 4) << 1;

    v8f acc = {};
#pragma unroll
    for (int kk = 0; kk < 256; kk += 4) {
        v2f a;
        a.x = ldsA[l15 * SA + kk + kh];
        a.y = ldsA[l15 * SA + kk + kh + 1];
        v2f b;
        b.x = ldsB[(kk + kh) * SB + n0 + l15];
        b.y = ldsB[(kk + kh + 1) * SB + n0 + l15];
        acc = __builtin_amdgcn_wmma_f32_16x16x4_f32(false, a, false, b,
                                                    (short)0, acc, false, false);
    }

    const int n    = n0 + l15;
    const float w256 = W1[256 * HH + n];
    const float w257 = W1[257 * HH + n];
    const float bv   = b1[n];
    const int mb = row0 + ((lane >> 4) << 3);
#pragma unroll
    for (int r = 0; r < 8; ++r) {
        int m = mb + r;
        float v = acc[r] + tf[m * 2 + 0] * w256 + tf[m * 2 + 1] * w257 + bv;
        y1[(size_t)m * HH + n] = fmaxf(v, 0.0f);
    }
}

// ---------------- final 64->1 projection ----------------
__global__ void final_proj(const float* __restrict__ y2, const float* __restrict__ W3,
                           const float* __restrict__ b3, float* __restrict__ out) {
    int p = blockIdx.x * blockDim.x + threadIdx.x;
    if (p < PP) {
        float s = b3[0];
        const float* row = y2 + (size_t)p * 64;
#pragma unroll 8
        for (int k = 0; k < 64; ++k) s += row[k] * W3[k];
        out[p] = s;
    }
}

// ---------------- host orchestration ----------------
extern "C" void kernel_launch(void* const* d_in, const int* in_sizes, int n_in,
                              void* d_out, int out_size, void* d_ws, size_t ws_size,
                              hipStream_t stream) {
    (void)in_sizes; (void)n_in; (void)out_size; (void)ws_size;

    const float* x    = (const float*)d_in[0];
    const int*   ei   = (const int*)d_in[1];
    const int*   eSrc = ei;
    const int*   eDst = ei + EE;
    const int*   sn   = (const int*)d_in[2];
    const int*   tn   = (const int*)d_in[3];
    const float* tf   = (const float*)d_in[4];
    const float* Wemb = (const float*)d_in[5];
    const float* bemb = (const float*)d_in[6];
    const float* Ws   = (const float*)d_in[7];
    const float* bs   = (const float*)d_in[8];
    const float* gam  = (const float*)d_in[9];
    const float* bet  = (const float*)d_in[10];
    const float* W1   = (const float*)d_in[11];
    const float* b1   = (const float*)d_in[12];
    const float* W2   = (const float*)d_in[13];
    const float* b2   = (const float*)d_in[14];
    const float* W3   = (const float*)d_in[15];
    const float* b3   = (const float*)d_in[16];
    float* out = (float*)d_out;

    uintptr_t base = (uintptr_t)d_ws;
    auto alloc = [&](size_t elems) -> float* {
        base = (base + 255) & ~(uintptr_t)255;
        float* p = (float*)base;
        base += elems * sizeof(float);
        return p;
    };
    float* h     = alloc((size_t)NN * HH);
    float* z     = alloc((size_t)NN * HH);
    float* hb    = alloc((size_t)NN * HH);
    float* deg   = alloc(NN);
    float* dinv  = alloc(NN);
    float* nself = alloc(NN);
    float* ne    = alloc(EE);
    float* stats = alloc(2 * HH);
    float* sums  = stats;
    float* sqs   = stats + HH;
    float* mu    = alloc(HH);
    float* rstd  = alloc(HH);
    float* y1    = alloc((size_t)PP * HH);
    float* y2    = alloc((size_t)PP * (HH / 2));

    // ---- degree / GCN normalization (once) ----
    (void)hipMemsetAsync(deg, 0, NN * sizeof(float), stream);
    deg_kernel<<<(EE + 255) / 256, 256, 0, stream>>>(eDst, deg);
    dinv_kernel<<<(NN + 255) / 256, 256, 0, stream>>>(deg, dinv, nself);
    norme_kernel<<<(EE + 255) / 256, 256, 0, stream>>>(eSrc, eDst, dinv, ne);

    // ---- embed: h = relu(x @ W_emb + b_emb), K=32 ----
    {
        constexpr size_t shb = (16 * (FIN + 4) + FIN * (HH + 8)) * sizeof(float);
        gemm_wmma<FIN, HH><<<NN / 16, 256, shb, stream>>>(x, Wemb, bemb, h, FIN, HH, 1);
    }

    // ---- 3 GCN layers ----
    for (int i = 0; i < LL; ++i) {
        constexpr size_t shz = (16 * (HH + 4) + HH * (HH + 8)) * sizeof(float);
        gemm_wmma<HH, HH><<<NN / 16, 256, shz, stream>>>(
            h, Ws + (size_t)i * HH * HH, nullptr, z, HH, HH, 0);
        init_h<<<(NN * HH) / 256, 256, 0, stream>>>(z, nself, bs + i * HH, hb);
        edge_agg<<<EE / 8, 256, 0, stream>>>(eSrc, eDst, ne, z, hb);
        (void)hipMemsetAsync(stats, 0, 2 * HH * sizeof(float), stream);
        bn_stats<<<256, 256, 0, stream>>>(hb, sums, sqs);
        bn_final<<<1, HH, 0, stream>>>(sums, sqs, mu, rstd);
        bn_apply<<<(NN * HH) / 256, 256, 0, stream>>>(
            hb, mu, rstd, gam + i * HH, bet + i * HH, (i < LL - 1) ? 1 : 0, h);
    }

    // ---- pair MLP ----
    {
        constexpr size_t shp = (16 * 260 + 256 * 136) * sizeof(float);
        pair_mlp1_wmma<<<PP / 16, 256, shp, stream>>>(h, sn, tn, tf, W1, b1, y1);
        constexpr size_t sh2 = (16 * (HH + 4) + HH * (HH / 2 + 8)) * sizeof(float);
        gemm_wmma<HH, HH / 2><<<PP / 16, 128, sh2, stream>>>(y1, W2, b2, y2, HH, HH / 2, 1);
    }
    final_proj<<<(PP + 255) / 256, 256, 0, stream>>>(y2, W3, b3, out);
}